// ConvexMixer_1039382086410
// MI455X (gfx1250) — compile-verified
//
#include <hip/hip_runtime.h>
#include <hip/hip_bf16.h>
#include <math.h>

#define B_  2
#define H_  8
#define S_  1024
#define D_  64
#define P_  8
#define R_  32
#define D1_ 128
#define NTOK (B_*H_*S_)
#define EPS_ 1e-6f
#define NU_MIXER_ 2.71828f
#define NU_HULL_  0.9999993f   /* log(2.71828) */

typedef __attribute__((ext_vector_type(2))) float v2f;
typedef __attribute__((ext_vector_type(8))) float v8f;

__device__ __forceinline__ v8f wmma4(v2f a, v2f b, v8f c) {
  // D = A(16x4 f32) * B(4x16 f32) + C(16x16 f32)
  return __builtin_amdgcn_wmma_f32_16x16x4_f32(false, a, false, b, (short)0, c, false, false);
}

// Branchless fast softplus: v_exp_f32 + v_add + v_log_f32 + v_cndmask.
__device__ __forceinline__ float softplusf(float x) {
  float l = __logf(1.f + __expf(x));
  return (x > 20.f) ? x : l;
}
__device__ __forceinline__ float sigmoidf(float x) {
  return 1.f / (1.f + __expf(-x));
}

// ---------------------------------------------------------------------------
// Kernel 1: fold hull params into GEMM-ready weights.
//   W0t[p][d=64][n=128] = softplus(wr0[p][n][d])^2 * sigmoid(g0[p])
//   b0f[p][n]           = b0[p][n] * sigmoid(g0[p])
//   W1c[p][k=128][d=64] = softplus(wr1[p][d][k])^2 * sigmoid(g1[p]) + zw[p][k][d]
//   b1f[p][d]           = b1[p][d] * sigmoid(g1[p])
// ---------------------------------------------------------------------------
__global__ void prep_hull_kernel(const float* __restrict__ wr0, const float* __restrict__ b0,
                                 const float* __restrict__ wr1, const float* __restrict__ b1,
                                 const float* __restrict__ zw,  const float* __restrict__ g0,
                                 const float* __restrict__ g1,
                                 float* __restrict__ W0t, float* __restrict__ b0f,
                                 float* __restrict__ W1c, float* __restrict__ b1f)
{
  int idx = blockIdx.x * blockDim.x + threadIdx.x;
  const int n_w0 = P_ * D_ * D1_;
  const int n_w1 = P_ * D1_ * D_;
  if (idx < n_w0) {
    int p = idx / (D_ * D1_); int rem = idx % (D_ * D1_);
    int d = rem / D1_;        int n = rem % D1_;
    float s = softplusf(wr0[(p * D1_ + n) * D_ + d]);
    W0t[idx] = s * s * sigmoidf(g0[p]);
  } else if (idx < n_w0 + n_w1) {
    int j = idx - n_w0;
    int p = j / (D1_ * D_);  int rem = j % (D1_ * D_);
    int k = rem / D_;        int d = rem % D_;
    float s = softplusf(wr1[(p * D_ + d) * D1_ + k]);
    W1c[j] = s * s * sigmoidf(g1[p]) + zw[(p * D1_ + k) * D_ + d];
  } else if (idx < n_w0 + n_w1 + P_ * D1_) {
    int j = idx - n_w0 - n_w1;
    b0f[j] = b0[j] * sigmoidf(g0[j / D1_]);
  } else if (idx < n_w0 + n_w1 + P_ * D1_ + P_ * D_) {
    int j = idx - n_w0 - n_w1 - P_ * D1_;
    b1f[j] = b1[j] * sigmoidf(g1[j / D_]);
  }
}

// ---------------------------------------------------------------------------
// Kernel 2: per-token hull evaluation (f = lse/tau), phi features, mixer tau.
// One block = 16 tokens, 4 waves; wave w handles prototypes w and w+4.
// GEMMs via V_WMMA_F32_16X16X4_F32.
// ---------------------------------------------------------------------------
__global__ __launch_bounds__(128) void hull_kernel(
    const float* __restrict__ X,
    const float* __restrict__ W0t, const float* __restrict__ b0f,
    const float* __restrict__ W1c, const float* __restrict__ b1f,
    const float* __restrict__ ob,
    const float* __restrict__ gw,  const float* __restrict__ gb,
    const float* __restrict__ wh,
    float* __restrict__ phi_out, float* __restrict__ f_out,
    float* __restrict__ taumix_out, int is_q)
{
  __shared__ float xs[16][68];        // raw x tile (padded: conflict-free cols)
  __shared__ float xg[16][68];        // gated x
  __shared__ float z0s[4][16][132];   // per-wave layer-0 activations (padded)
  __shared__ float taus[16];
  __shared__ float scores_s[P_][16];

  const int tid  = threadIdx.x;
  const int tok0 = blockIdx.x * 16;

  for (int i = tid; i < 16 * 64; i += 128)
    xs[i >> 6][i & 63] = X[(tok0 + (i >> 6)) * D_ + (i & 63)];
  __syncthreads();

  if (tid < 16) {
    int t = tid;
    float dot = gb[0];
    for (int d = 0; d < 64; ++d) dot += xs[t][d] * gw[d];
    float g = 1.f - __expf(-softplusf(dot));
    float s2 = 0.f;
    for (int d = 0; d < 64; ++d) { float v = xs[t][d] * g; xg[t][d] = v; s2 += v * v; }
    taus[t] = sqrtf(s2 * (1.f / 64.f) + EPS_ + NU_HULL_);
    if (is_q) {
      float qs = 0.f;
      for (int d = 0; d < 64; ++d) {
        float qv = xs[t][d]; float qp = qv * softplusf(qv); qs += qp * qp;
      }
      taumix_out[tok0 + t] = sqrtf(qs * (1.f / 64.f) + EPS_ + NU_MIXER_);
    }
  }
  // phi = softplus(min(x @ wh^T, 20))
  for (int i = tid; i < 16 * R_; i += 128) {
    int t = i / R_; int r = i % R_;
    float acc = 0.f;
    for (int d = 0; d < 64; ++d) acc += xs[t][d] * wh[r * D_ + d];
    phi_out[(tok0 + t) * R_ + r] = softplusf(fminf(acc, 20.f));
  }
  __syncthreads();

  const int lane  = tid & 31;
  const int wv    = tid >> 5;
  const int m     = lane & 15;
  const int koff  = (lane < 16) ? 0 : 2;
  const int rbase = (lane < 16) ? 0 : 8;

  // hoist A-fragments for layer 0 (K = 64 -> 16 chunks of 4)
  v2f a0[16];
  #pragma unroll
  for (int j = 0; j < 16; ++j) {
    a0[j][0] = xg[m][4 * j + koff];
    a0[j][1] = xg[m][4 * j + koff + 1];
  }

  for (int pp = 0; pp < 2; ++pp) {
    int p = wv + pp * 4;
    // ---- layer 0: z0 = softplus(xg @ W0t[p] + b0f[p]), 16x128 ----
    #pragma unroll
    for (int nt = 0; nt < 8; ++nt) {
      v8f c = {};
      const float* wp = W0t + (size_t)(p * 64 + koff) * D1_ + nt * 16 + m;
      #pragma unroll
      for (int j = 0; j < 16; ++j) {
        v2f b; b[0] = wp[4 * j * D1_]; b[1] = wp[(4 * j + 1) * D1_];
        c = wmma4(a0[j], b, c);
      }
      float bias = b0f[p * D1_ + nt * 16 + m];
      #pragma unroll
      for (int i = 0; i < 8; ++i)
        z0s[wv][rbase + i][nt * 16 + m] = softplusf(c[i] + bias);
    }
    __syncthreads();
    // ---- layer 1: out = softplus(z0 @ W1c[p] + b1f[p]) + ob[p], mean over d ----
    v2f a1[32];
    #pragma unroll
    for (int j = 0; j < 32; ++j) {
      a1[j][0] = z0s[wv][m][4 * j + koff];
      a1[j][1] = z0s[wv][m][4 * j + koff + 1];
    }
    float rowacc[8];
    #pragma unroll
    for (int i = 0; i < 8; ++i) rowacc[i] = 0.f;
    #pragma unroll
    for (int nt = 0; nt < 4; ++nt) {
      v8f c = {};
      const float* wp = W1c + (size_t)(p * D1_ + koff) * D_ + nt * 16 + m;
      #pragma unroll
      for (int j = 0; j < 32; ++j) {
        v2f b; b[0] = wp[4 * j * D_]; b[1] = wp[(4 * j + 1) * D_];
        c = wmma4(a1[j], b, c);
      }
      int n = nt * 16 + m;
      float bias = b1f[p * D_ + n];
      float obv  = ob[p * D_ + n];
      #pragma unroll
      for (int i = 0; i < 8; ++i)
        rowacc[i] += softplusf(c[i] + bias) + obv;
    }
    #pragma unroll
    for (int i = 0; i < 8; ++i) {
      float v = rowacc[i];
      v += __shfl_xor(v, 1, 32);
      v += __shfl_xor(v, 2, 32);
      v += __shfl_xor(v, 4, 32);
      v += __shfl_xor(v, 8, 32);
      rowacc[i] = v;
    }
    if (m == 0) {
      #pragma unroll
      for (int i = 0; i < 8; ++i)
        scores_s[p][rbase + i] = rowacc[i] * (1.f / 64.f);
    }
    __syncthreads();
  }

  if (tid < 16) {
    int t = tid;
    float tau = taus[t];
    float mx = -3.0e38f;
    #pragma unroll
    for (int p = 0; p < P_; ++p) mx = fmaxf(mx, scores_s[p][t] * tau);
    float s = 0.f;
    #pragma unroll
    for (int p = 0; p < P_; ++p) s += __expf(scores_s[p][t] * tau - mx);
    f_out[tok0 + t] = (mx + __logf(s)) / tau;
  }
}

// ---------------------------------------------------------------------------
// Kernel 3: flash-style causal attention. One wave per 16-row query tile.
// S = phi_q.phi_k^T (WMMA) -> log + biases -> online softmax -> P.V (WMMA).
// extra_score streamed exactly once.
// ---------------------------------------------------------------------------
__global__ __launch_bounds__(32) void attn_kernel(
    const float* __restrict__ phi_q, const float* __restrict__ phi_k,
    const float* __restrict__ fq,    const float* __restrict__ gk,
    const float* __restrict__ taumix, const float* __restrict__ V,
    const float* __restrict__ extra, float* __restrict__ out)
{
  __shared__ float ps[16][20];   // P-tile staging (padded: conflict-free)
  const int lane  = threadIdx.x & 31;
  const int m     = lane & 15;
  const int koff  = (lane < 16) ? 0 : 2;
  const int rbase = (lane < 16) ? 0 : 8;
  const int q0 = blockIdx.x * 16;
  const int bh = blockIdx.y;
  const int tb = bh * S_;

  v2f aq[8];
  #pragma unroll
  for (int j = 0; j < 8; ++j) {
    const float* pq = phi_q + (size_t)(tb + q0 + m) * R_ + 4 * j + koff;
    aq[j][0] = pq[0]; aq[j][1] = pq[1];
  }
  float fqr[8], taur[8], mrow[8], lrow[8];
  #pragma unroll
  for (int i = 0; i < 8; ++i) {
    int row = q0 + rbase + i;
    fqr[i]  = fq[tb + row];
    taur[i] = taumix[tb + row];
    mrow[i] = -3.0e38f; lrow[i] = 0.f;
  }
  v8f o0 = {}, o1 = {}, o2 = {}, o3 = {};

  for (int t0 = 0; t0 <= q0; t0 += 16) {
    // S-tile = phi_q(16x32) . phi_k^T(32x16)
    v8f c = {};
    const float* pk = phi_k + (size_t)(tb + t0 + m) * R_;
    #pragma unroll
    for (int j = 0; j < 8; ++j) {
      v2f b; b[0] = pk[4 * j + koff]; b[1] = pk[4 * j + koff + 1];
      c = wmma4(aq[j], b, c);
    }
    float gkv = gk[tb + t0 + m];
    float st[8];
    #pragma unroll
    for (int i = 0; i < 8; ++i) {
      int qrow = q0 + rbase + i;
      int key  = t0 + m;
      float sc = __logf(c[i] + EPS_) + fqr[i] + gkv
               + extra[((size_t)bh * S_ + qrow) * S_ + key];
      st[i] = (key <= qrow) ? sc * taur[i] : -3.0e38f;
    }
    // online softmax update (row = i + rbase; cols spread across 16 lanes)
    #pragma unroll
    for (int i = 0; i < 8; ++i) {
      float v = st[i];
      v = fmaxf(v, __shfl_xor(v, 1, 32));
      v = fmaxf(v, __shfl_xor(v, 2, 32));
      v = fmaxf(v, __shfl_xor(v, 4, 32));
      v = fmaxf(v, __shfl_xor(v, 8, 32));
      float mnew = fmaxf(mrow[i], v);
      float corr = __expf(mrow[i] - mnew);
      float pval = __expf(st[i] - mnew);      // masked entries -> exp(-huge)=0
      float rs = pval;
      rs += __shfl_xor(rs, 1, 32);
      rs += __shfl_xor(rs, 2, 32);
      rs += __shfl_xor(rs, 4, 32);
      rs += __shfl_xor(rs, 8, 32);
      lrow[i] = lrow[i] * corr + rs;
      mrow[i] = mnew;
      o0[i] *= corr; o1[i] *= corr; o2[i] *= corr; o3[i] *= corr;
      ps[rbase + i][m] = pval;
    }
    __syncthreads();
    // O += P(16x16) . V(16x64)
    v2f ap[4];
    #pragma unroll
    for (int j = 0; j < 4; ++j) {
      ap[j][0] = ps[m][4 * j + koff];
      ap[j][1] = ps[m][4 * j + koff + 1];
    }
    const float* vb = V + (size_t)(tb + t0) * D_;
    #pragma unroll
    for (int j = 0; j < 4; ++j) {
      int k0 = 4 * j + koff;
      v2f b;
      b[0] = vb[k0 * D_ +  0 + m]; b[1] = vb[(k0 + 1) * D_ +  0 + m]; o0 = wmma4(ap[j], b, o0);
      b[0] = vb[k0 * D_ + 16 + m]; b[1] = vb[(k0 + 1) * D_ + 16 + m]; o1 = wmma4(ap[j], b, o1);
      b[0] = vb[k0 * D_ + 32 + m]; b[1] = vb[(k0 + 1) * D_ + 32 + m]; o2 = wmma4(ap[j], b, o2);
      b[0] = vb[k0 * D_ + 48 + m]; b[1] = vb[(k0 + 1) * D_ + 48 + m]; o3 = wmma4(ap[j], b, o3);
    }
    __syncthreads();
  }
  #pragma unroll
  for (int i = 0; i < 8; ++i) {
    int row = q0 + rbase + i;
    float inv = 1.f / fmaxf(lrow[i], EPS_);
    float* op = out + (size_t)(tb + row) * D_;
    op[m]      = o0[i] * inv;
    op[16 + m] = o1[i] * inv;
    op[32 + m] = o2[i] * inv;
    op[48 + m] = o3[i] * inv;
  }
}

// ---------------------------------------------------------------------------
extern "C" void kernel_launch(void* const* d_in, const int* in_sizes, int n_in,
                              void* d_out, int out_size, void* d_ws, size_t ws_size,
                              hipStream_t stream) {
  (void)in_sizes; (void)n_in; (void)out_size; (void)ws_size;
  const float* q     = (const float*)d_in[0];
  const float* k     = (const float*)d_in[1];
  const float* v     = (const float*)d_in[2];
  const float* extra = (const float*)d_in[3];
  // d_in[4] = mask (causal tril) — computed analytically in-kernel.
  const float* q_wr0 = (const float*)d_in[5];
  const float* q_b0  = (const float*)d_in[6];
  const float* q_wr1 = (const float*)d_in[7];
  const float* q_b1  = (const float*)d_in[8];
  const float* q_zw  = (const float*)d_in[9];
  const float* q_g0  = (const float*)d_in[10];
  const float* q_g1  = (const float*)d_in[11];
  const float* q_ob  = (const float*)d_in[12];
  const float* q_gw  = (const float*)d_in[13];
  const float* q_gb  = (const float*)d_in[14];
  const float* k_wr0 = (const float*)d_in[15];
  const float* k_b0  = (const float*)d_in[16];
  const float* k_wr1 = (const float*)d_in[17];
  const float* k_b1  = (const float*)d_in[18];
  const float* k_zw  = (const float*)d_in[19];
  const float* k_g0  = (const float*)d_in[20];
  const float* k_g1  = (const float*)d_in[21];
  const float* k_ob  = (const float*)d_in[22];
  const float* k_gw  = (const float*)d_in[23];
  const float* k_gb  = (const float*)d_in[24];
  const float* whq   = (const float*)d_in[25];
  const float* whk   = (const float*)d_in[26];

  float* ws = (float*)d_ws;
  size_t off = 0;
  auto alloc = [&](size_t n) { float* p = ws + off; off += n; return p; };
  float* W0t_q = alloc(P_ * D_ * D1_);
  float* W1c_q = alloc(P_ * D1_ * D_);
  float* b0f_q = alloc(P_ * D1_);
  float* b1f_q = alloc(P_ * D_);
  float* W0t_k = alloc(P_ * D_ * D1_);
  float* W1c_k = alloc(P_ * D1_ * D_);
  float* b0f_k = alloc(P_ * D1_);
  float* b1f_k = alloc(P_ * D_);
  float* phi_q = alloc((size_t)NTOK * R_);
  float* phi_k = alloc((size_t)NTOK * R_);
  float* fq_w  = alloc(NTOK);
  float* gk_w  = alloc(NTOK);
  float* tau_w = alloc(NTOK);

  const int prep_total = P_ * D_ * D1_ + P_ * D1_ * D_ + P_ * D1_ + P_ * D_;
  dim3 pgrid((prep_total + 255) / 256);
  prep_hull_kernel<<<pgrid, 256, 0, stream>>>(q_wr0, q_b0, q_wr1, q_b1, q_zw, q_g0, q_g1,
                                              W0t_q, b0f_q, W1c_q, b1f_q);
  prep_hull_kernel<<<pgrid, 256, 0, stream>>>(k_wr0, k_b0, k_wr1, k_b1, k_zw, k_g0, k_g1,
                                              W0t_k, b0f_k, W1c_k, b1f_k);

  hull_kernel<<<NTOK / 16, 128, 0, stream>>>(q, W0t_q, b0f_q, W1c_q, b1f_q, q_ob,
                                             q_gw, q_gb, whq, phi_q, fq_w, tau_w, 1);
  hull_kernel<<<NTOK / 16, 128, 0, stream>>>(k, W0t_k, b0f_k, W1c_k, b1f_k, k_ob,
                                             k_gw, k_gb, whk, phi_k, gk_w, tau_w, 0);

  attn_kernel<<<dim3(S_ / 16, B_ * H_), 32, 0, stream>>>(phi_q, phi_k, fq_w, gk_w,
                                                         tau_w, v, extra, (float*)d_out);
}